// PillarVoxelNet_19318762898070
// MI455X (gfx1250) — compile-verified
//
#include <hip/hip_runtime.h>
#include <cstdint>

// ---------------- problem constants (match reference) ----------------
#define GX        512
#define GY        512
#define NBATCH    4
#define NKEYS     (NBATCH * GX * GY)      // 2^20 == MAX_VOXELS
#define MAXVOX    NKEYS

#define TILE_PTS   128                    // points per TDM tile
#define TILE_ELEMS (TILE_PTS * 5)         // 640 f32 elements, 2560 B
#define WAVES_PB   8                      // 256 threads / wave32

typedef unsigned int u32x4 __attribute__((ext_vector_type(4)));
typedef int          i32x8 __attribute__((ext_vector_type(8)));
typedef int          i32x4 __attribute__((ext_vector_type(4)));

// ---------------- TDM: 1D tile (640 x f32) global -> LDS ----------------
__device__ __forceinline__ void tdm_load_tile(unsigned lds_byte_addr, const float* gsrc) {
    unsigned long long ga = (unsigned long long)(uintptr_t)gsrc;
    // D# group 0 (ISA 8.3): count=1 | lds_addr | global_addr[56:0] | type=2
    u32x4 g0 = { 1u,
                 lds_byte_addr,
                 (unsigned)(ga & 0xffffffffull),
                 (unsigned)((ga >> 32) & 0x1ffffffull) | (2u << 30) };
    // D# group 1 (ISA 8.4): data_size=2 (4B); tensor_dim0=tile_dim0=640;
    // tensor_dim1=1, tile_dim1=1; dim0_stride=640; dim1_stride=640.
    i32x8 g1 = { (int)(2u << 16),                       // workgroup_mask=0, data_size=4B
                 (int)((TILE_ELEMS & 0xffffu) << 16),   // tensor_dim0[15:0] @ bits 63:48
                 (int)((TILE_ELEMS >> 16) | (1u << 16)),// tensor_dim0[31:16], tensor_dim1[15:0]=1
                 (int)(0u | ((unsigned)TILE_ELEMS << 16)), // tensor_dim1[31:16]=0, tile_dim0=640
                 (int)1,                                // tile_dim1=1, tile_dim2=0
                 (int)TILE_ELEMS,                       // tensor_dim0_stride[31:0]
                 (int)((unsigned)TILE_ELEMS << 16),     // stride0[47:32]=0, dim1_stride[15:0]
                 0 };
    i32x4 gz4 = { 0, 0, 0, 0 };
    i32x8 gz8 = { 0, 0, 0, 0, 0, 0, 0, 0 };
    __builtin_amdgcn_tensor_load_to_lds(g0, g1, gz4, gz4, gz8, 0);
}

__device__ __forceinline__ int clamp_coord(float v, float lo) {
    // reference: clip((v - lo) / 0.2, 0, 511) then truncate
    float t = (v - lo) / 0.2f;
    t = fminf(fmaxf(t, 0.0f), 511.0f);
    return (int)t;
}

__device__ __forceinline__ unsigned make_key(float b, float x, float y) {
    int cx = clamp_coord(x, -51.2f);
    int cy = clamp_coord(y, -51.2f);
    int ib = (int)b;
    return ((unsigned)ib << 18) | ((unsigned)cx << 9) | (unsigned)cy;
}

// ---------------- kernel 1: scatter-accumulate sums/count ----------------
__global__ __launch_bounds__(256) void pv_accum(const float* __restrict__ pts,
                                                float* __restrict__ acc,  // NKEYS * 4 floats
                                                int n) {
    __shared__ float smem[WAVES_PB * 2 * TILE_ELEMS];   // 40 KB
    const int lane  = threadIdx.x & 31;
    const int wave  = threadIdx.x >> 5;
    const long long ntiles = n / TILE_PTS;
    const long long wstride = (long long)gridDim.x * WAVES_PB;
    long long w = (long long)blockIdx.x * WAVES_PB + wave;

    float* wbuf = &smem[wave * 2 * TILE_ELEMS];
    unsigned ldsBase = (unsigned)(uintptr_t)(void*)wbuf;  // low 32 bits = LDS byte offset

    long long t = w;
    int parity = 0;
    if (t < ntiles) tdm_load_tile(ldsBase, pts + t * (long long)TILE_ELEMS);

    for (; t < ntiles; t += wstride) {
        long long tn = t + wstride;
        if (tn < ntiles) {
            asm volatile("s_wait_dscnt 0" ::: "memory");  // prior reads of target buffer done
            tdm_load_tile(ldsBase + (unsigned)((parity ^ 1) * TILE_ELEMS * 4),
                          pts + tn * (long long)TILE_ELEMS);
            __builtin_amdgcn_s_wait_tensorcnt(1);         // current tile's DMA complete
        } else {
            __builtin_amdgcn_s_wait_tensorcnt(0);
        }
        asm volatile("" ::: "memory");                    // LDS was written by TDM
        const float* buf = wbuf + parity * TILE_ELEMS;
        #pragma unroll
        for (int j = 0; j < TILE_PTS / 32; ++j) {
            int idx = j * 32 + lane;
            float b = buf[idx * 5 + 0];
            float x = buf[idx * 5 + 1];
            float y = buf[idx * 5 + 2];
            float z = buf[idx * 5 + 3];
            unsigned key = make_key(b, x, y);
            float* a = acc + (size_t)key * 4;
            atomicAdd(a + 0, x);
            atomicAdd(a + 1, y);
            atomicAdd(a + 2, z);
            atomicAdd(a + 3, 1.0f);
        }
        parity ^= 1;
    }

    // remainder points (n not multiple of TILE_PTS) via direct loads
    int rem = n - (int)(ntiles * TILE_PTS);
    if (rem > 0 && blockIdx.x == 0 && threadIdx.x < rem) {
        size_t p = (size_t)ntiles * TILE_PTS + threadIdx.x;
        float b = pts[p * 5 + 0], x = pts[p * 5 + 1];
        float y = pts[p * 5 + 2], z = pts[p * 5 + 3];
        unsigned key = make_key(b, x, y);
        float* a = acc + (size_t)key * 4;
        atomicAdd(a + 0, x); atomicAdd(a + 1, y);
        atomicAdd(a + 2, z); atomicAdd(a + 3, 1.0f);
    }
}

// ---------------- kernel 2: per-block exclusive scan of occupancy ----------------
__global__ __launch_bounds__(256) void pv_scan1(const float* __restrict__ acc,
                                                unsigned* __restrict__ localPfx,
                                                unsigned* __restrict__ blockSums) {
    __shared__ unsigned tmp[256];
    const int tid = threadIdx.x;
    const unsigned base = blockIdx.x * 1024u + tid * 4u;
    unsigned f0 = acc[(size_t)(base + 0) * 4 + 3] > 0.f ? 1u : 0u;
    unsigned f1 = acc[(size_t)(base + 1) * 4 + 3] > 0.f ? 1u : 0u;
    unsigned f2 = acc[(size_t)(base + 2) * 4 + 3] > 0.f ? 1u : 0u;
    unsigned f3 = acc[(size_t)(base + 3) * 4 + 3] > 0.f ? 1u : 0u;
    unsigned s = f0 + f1 + f2 + f3;
    tmp[tid] = s;
    __syncthreads();
    #pragma unroll
    for (int off = 1; off < 256; off <<= 1) {
        unsigned v = (tid >= off) ? tmp[tid - off] : 0u;
        __syncthreads();
        tmp[tid] += v;
        __syncthreads();
    }
    unsigned excl = tmp[tid] - s;
    localPfx[base + 0] = excl;
    localPfx[base + 1] = excl + f0;
    localPfx[base + 2] = excl + f0 + f1;
    localPfx[base + 3] = excl + f0 + f1 + f2;
    if (tid == 255) blockSums[blockIdx.x] = tmp[tid];   // inclusive block total
}

// ---------------- kernel 3: exclusive scan of 1024 block totals ----------------
__global__ __launch_bounds__(1024) void pv_scan2(unsigned* __restrict__ blockSums) {
    __shared__ unsigned tmp[1024];
    const int tid = threadIdx.x;
    unsigned v = blockSums[tid];
    tmp[tid] = v;
    __syncthreads();
    #pragma unroll
    for (int off = 1; off < 1024; off <<= 1) {
        unsigned u = (tid >= off) ? tmp[tid - off] : 0u;
        __syncthreads();
        tmp[tid] += u;
        __syncthreads();
    }
    blockSums[tid] = tmp[tid] - v;   // exclusive
}

// ---------------- kernel 4: scatter unique voxel coords (b, cy, cx) ----------------
__global__ __launch_bounds__(256) void pv_unq(const float* __restrict__ acc,
                                              const unsigned* __restrict__ localPfx,
                                              const unsigned* __restrict__ blockOff,
                                              float* __restrict__ unqOut,   // MAXVOX*3 (pre-zeroed)
                                              float* __restrict__ gridOut) {
    unsigned k = blockIdx.x * blockDim.x + threadIdx.x;
    if (k == 0) { gridOut[0] = (float)GY; gridOut[1] = (float)GX; }
    if (k >= NKEYS) return;
    if (acc[(size_t)k * 4 + 3] > 0.f) {
        unsigned r = localPfx[k] + blockOff[k >> 10];
        float* o = unqOut + (size_t)r * 3;
        o[0] = (float)(k >> 18);          // b
        o[1] = (float)(k & 511u);         // cy
        o[2] = (float)((k >> 9) & 511u);  // cx
    }
}

// ---------------- kernel 5: per-point 9 features + inverse index ----------------
__global__ __launch_bounds__(256) void pv_feat(const float* __restrict__ pts,
                                               const float4* __restrict__ acc,
                                               const unsigned* __restrict__ localPfx,
                                               const unsigned* __restrict__ blockOff,
                                               float* __restrict__ featOut,  // n*9
                                               float* __restrict__ invOut,   // n
                                               int n) {
    long long gsz = (long long)gridDim.x * blockDim.x;
    for (long long p = (long long)blockIdx.x * blockDim.x + threadIdx.x; p < n; p += gsz) {
        if (p + gsz < n) __builtin_prefetch(&pts[(size_t)(p + gsz) * 5], 0, 1);
        const float* pp = pts + (size_t)p * 5;
        float b = pp[0], x = pp[1], y = pp[2], z = pp[3], inten = pp[4];
        int cx = clamp_coord(x, -51.2f);
        int cy = clamp_coord(y, -51.2f);
        unsigned key = ((unsigned)(int)b << 18) | ((unsigned)cx << 9) | (unsigned)cy;
        float4 a = acc[key];
        float cnt = fmaxf(a.w, 1.0f);
        float mx = a.x / cnt, my = a.y / cnt, mz = a.z / cnt;
        float ctx = ((float)cx * 0.2f + 0.1f) + (-51.2f);
        float cty = ((float)cy * 0.2f + 0.1f) + (-51.2f);
        float* o = featOut + (size_t)p * 9;
        o[0] = x;        o[1] = y;        o[2] = z;  o[3] = inten;
        o[4] = x - mx;   o[5] = y - my;   o[6] = z - mz;
        o[7] = x - ctx;  o[8] = y - cty;
        invOut[p] = (float)(localPfx[key] + blockOff[key >> 10]);
    }
}

// ---------------- launch ----------------
extern "C" void kernel_launch(void* const* d_in, const int* in_sizes, int n_in,
                              void* d_out, int out_size, void* d_ws, size_t ws_size,
                              hipStream_t stream) {
    const float* pts = (const float*)d_in[0];
    const int n = in_sizes[0] / 5;

    // output layout: features[n*9] | unq[MAXVOX*3] | unq_inv[n] | grid_yx[2]
    float* out      = (float*)d_out;
    float* featOut  = out;
    float* unqOut   = out + (size_t)n * 9;
    float* invOut   = unqOut + (size_t)MAXVOX * 3;
    float* gridOut  = invOut + (size_t)n;

    // workspace layout
    char*     ws        = (char*)d_ws;
    float*    acc       = (float*)ws;                                  // NKEYS * float4 = 16 MB
    unsigned* localPfx  = (unsigned*)(ws + (size_t)NKEYS * 16);        // 4 MB
    unsigned* blockSums = (unsigned*)(ws + (size_t)NKEYS * 16 + (size_t)NKEYS * 4);  // 4 KB

    (void)hipMemsetAsync(acc, 0, (size_t)NKEYS * 16, stream);
    (void)hipMemsetAsync(unqOut, 0, (size_t)MAXVOX * 3 * sizeof(float), stream);  // fill_value=0

    pv_accum<<<512, 256, 0, stream>>>(pts, acc, n);
    pv_scan1<<<NKEYS / 1024, 256, 0, stream>>>(acc, localPfx, blockSums);
    pv_scan2<<<1, 1024, 0, stream>>>(blockSums);
    pv_unq<<<NKEYS / 256, 256, 0, stream>>>(acc, localPfx, blockSums, unqOut, gridOut);
    pv_feat<<<4096, 256, 0, stream>>>(pts, (const float4*)acc, localPfx, blockSums,
                                      featOut, invOut, n);
}